// MultiModelNetV2_49744311222531
// MI455X (gfx1250) — compile-verified
//
#include <hip/hip_runtime.h>
#include <math.h>
#include <stdint.h>

// ---------------------------------------------------------------------------
// MultiModelNetV2 for MI455X (gfx1250, wave32).
// Dense GEMMs -> v_wmma_f32_16x16x4_f32 (full fp32, matches reference dtype),
// with the B (weight) panel staged into LDS per workgroup via CDNA5
// global_load_async_to_lds_b128 (ASYNCcnt) and 4-wide N-tile register
// blocking per wave (one A fragment feeds 4 WMMAs, B fragments from LDS).
// Edge softmax/scatter -> VALU + global atomics. LN/attn reductions -> wave32
// shfl butterflies. Input pytree flattened in setup_inputs() insertion order.
// ---------------------------------------------------------------------------

#define NTOT   68000L
#define E_PER  50000
#define NPROTO 10

typedef float v2f __attribute__((ext_vector_type(2)));
typedef float v8f __attribute__((ext_vector_type(8)));

__device__ __forceinline__ float wsum(float v) {
#pragma unroll
  for (int o = 16; o > 0; o >>= 1) v += __shfl_xor(v, o, 32);
  return v;
}
__device__ __forceinline__ unsigned flipf(float f) {
  unsigned u = __float_as_uint(f);
  return (u & 0x80000000u) ? ~u : (u | 0x80000000u);
}
__device__ __forceinline__ float unflipf(unsigned u) {
  return (u & 0x80000000u) ? __uint_as_float(u ^ 0x80000000u) : __uint_as_float(~u);
}

// CDNA5 async global->LDS copy (16B per lane), tracked with ASYNCcnt.
// vdst = wave-relative LDS byte offset (low 32 bits of generic shared ptr:
// ISA flat->LDS aperture truncates addr[31:0]).
__device__ __forceinline__ void async_copy_b128(unsigned lds_off,
                                                const void* gptr) {
  asm volatile("global_load_async_to_lds_b128 %0, %1, off"
               :
               : "v"(lds_off), "v"((unsigned long long)(uintptr_t)gptr)
               : "memory");
}
__device__ __forceinline__ void wait_asynccnt0() {
  asm volatile("s_wait_asynccnt 0x0" ::: "memory");
}

// ---------------------------------------------------------------------------
// WMMA fp32 GEMM: C[M,N] = act(A[M,K] @ B + bias).
// B panel ([K][N] f32, K*N*4 <= 320KB LDS) staged to LDS once per block:
//   contiguous (!TRANSB && ldb==N): async b128 copies + s_wait_asynccnt
//   else: scalar gather/transpose stage via ds_store.
// One wave computes a 16 x (NB*16) output strip: A frag (16x4) loaded once
// per K-step, NB B frags read from LDS, NB wmma's.
// A frag: lane<16 -> row=rt*16+lane, K=k0,k0+1; lane>=16 -> K=k0+2,k0+3.
// B frag: lane<16 -> col n, K=k0,k0+1; upper half K=k0+2,k0+3.
// D: VGPR j holds M = rt*16 + 8*(lane>=16) + j.
// ---------------------------------------------------------------------------
template <int TRANSB, int ACT, int NB>  // ACT: 0 none, 1 relu, 2 sigmoid
__global__ __launch_bounds__(256) void gemm_wmma_kernel(
    const float* __restrict__ A, const int* __restrict__ gather,
    const float* __restrict__ B, const float* __restrict__ bias,
    float* __restrict__ C, int M, int N, int K, int lda, int ldb, int ldc) {
  extern __shared__ float bs[];  // [K][N]
  const int tid = threadIdx.x;
  const long bel = (long)K * N;

  if (!TRANSB && ldb == N) {
    const unsigned lds0 = (unsigned)(uintptr_t)bs;
    const long nchunk = bel >> 2;  // 16B chunks (N % 16 == 0 always)
    for (long c = tid; c < nchunk; c += 256)
      async_copy_b128(lds0 + (unsigned)(c << 4), (const char*)B + (c << 4));
    wait_asynccnt0();
  } else {
    for (long i = tid; i < bel; i += 256) {
      long k = i / N;
      int n = (int)(i - k * N);
      bs[i] = TRANSB ? B[(long)n * ldb + k] : B[k * (long)ldb + n];
    }
  }
  __syncthreads();

  const int tiles_n = N >> 4;
  const int groups_n = tiles_n / NB;
  const int total = (M >> 4) * groups_n;
  const int w = blockIdx.x * 8 + (tid >> 5);
  if (w >= total) return;  // wave-uniform (EXEC all-1 for WMMA)
  const int rt = w / groups_n, cg = w % groups_n;
  const int lane = tid & 31, hi = lane >> 4, lo = lane & 15;

  const int am = rt * 16 + lo;
  const long arow = gather ? (long)gather[am] : (long)am;
  const float* Ap = A + arow * (long)lda + hi * 2;
  const int n0 = cg * NB * 16 + lo;

  v8f acc[NB];
#pragma unroll
  for (int j = 0; j < NB; j++) acc[j] = v8f{};

  for (int k0 = 0; k0 < K; k0 += 4) {
    v2f a;
    a.x = Ap[k0];
    a.y = Ap[k0 + 1];
    const float* b0 = bs + (long)(k0 + hi * 2) * N + n0;
#pragma unroll
    for (int j = 0; j < NB; j++) {
      v2f b;
      b.x = b0[j * 16];
      b.y = b0[N + j * 16];
      acc[j] = __builtin_amdgcn_wmma_f32_16x16x4_f32(false, a, false, b,
                                                     (short)0, acc[j], false,
                                                     false);
    }
  }
  const int mbase = rt * 16 + hi * 8;
#pragma unroll
  for (int j = 0; j < NB; j++) {
    const int col = cg * NB * 16 + j * 16 + lo;
    const float bv = bias ? bias[col] : 0.f;
#pragma unroll
    for (int i = 0; i < 8; i++) {
      float v = acc[j][i] + bv;
      if (ACT == 1) v = fmaxf(v, 0.f);
      if (ACT == 2) v = 1.f / (1.f + expf(-v));
      C[(long)(mbase + i) * ldc + col] = v;
    }
  }
}

// ---------------------------------------------------------------------------
__global__ void fill_u32_kernel(unsigned* p, unsigned v, long n) {
  long i = (long)blockIdx.x * 256 + threadIdx.x;
  if (i < n) p[i] = v;
}

__global__ void copy_strided_kernel(float* dst, long dstride, const float* src,
                                    long sstride, long rows, int cols) {
  long i = (long)blockIdx.x * 256 + threadIdx.x;
  long r = i / cols;
  int c = (int)(i - r * cols);
  if (r < rows) dst[r * dstride + c] = src[r * sstride + c];
}

__global__ void gelu_kernel(float* p, long n) {
  long i = (long)blockIdx.x * 256 + threadIdx.x;
  if (i < n) {
    float x = p[i];
    p[i] = 0.5f * x * (1.f + erff(x * 0.70710678118654752f));
  }
}

// alpha[e,h] = scale * p_rel[h] * dot(q[dst[e],h,:], k_rel[e,h,:]); segment max
__global__ void alpha_max_kernel(const float* __restrict__ q,
                                 const float* __restrict__ krel,
                                 const int* __restrict__ didx,
                                 const float* __restrict__ p_rel, unsigned* mx,
                                 float* alpha, int E) {
  int t = blockIdx.x * 256 + threadIdx.x;
  if (t >= E * 4) return;
  int e = t >> 2, h = t & 3;
  int d = didx[e];
  const float* qr = q + (long)d * 128 + h * 32;
  const float* kr = krel + (long)e * 128 + h * 32;
  float s = 0.f;
#pragma unroll
  for (int i = 0; i < 32; i++) s += qr[i] * kr[i];
  s *= p_rel[h] * 0.17677669529663687f;  // 1/sqrt(32)
  alpha[t] = s;
  atomicMax(&mx[(long)d * 4 + h], flipf(s));
}

__global__ void softnorm_kernel(float* alpha, const unsigned* __restrict__ mx,
                                float* den, const int* __restrict__ didx, int E) {
  int t = blockIdx.x * 256 + threadIdx.x;
  if (t >= E * 4) return;
  int e = t >> 2, h = t & 3;
  int d = didx[e];
  float ex = expf(alpha[t] - unflipf(mx[(long)d * 4 + h]));
  alpha[t] = ex;
  atomicAdd(&den[(long)d * 4 + h], ex);
}

__global__ void scatter_kernel(const float* __restrict__ ex,
                               const float* __restrict__ den,
                               const float* __restrict__ vrel,
                               const int* __restrict__ didx, float* out, int E) {
  long t = (long)blockIdx.x * 256 + threadIdx.x;
  if (t >= (long)E * 128) return;
  int e = (int)(t >> 7);
  int c = (int)(t & 127);
  int h = c >> 5;
  int d = didx[e];
  float w = ex[(long)e * 4 + h] / (den[(long)d * 4 + h] + 1e-16f);
  atomicAdd(&out[(long)d * 128 + c], w * vrel[(long)e * 128 + c]);
}

// y = relu(LN(sig(skip)*o + (1-sig(skip))*x + res)), row length 128, wave/row
__global__ void epilogue_kernel(const float* __restrict__ o,
                                const float* __restrict__ x,
                                const float* __restrict__ res,
                                const float* __restrict__ skip,
                                const float* __restrict__ g,
                                const float* __restrict__ bb, float* y, int rows) {
  int w = (blockIdx.x * 256 + threadIdx.x) >> 5;
  int lane = threadIdx.x & 31;
  if (w >= rows) return;
  float a = 1.f / (1.f + expf(-skip[0]));
  float t[4];
#pragma unroll
  for (int j = 0; j < 4; j++) {
    long idx = (long)w * 128 + j * 32 + lane;
    t[j] = a * o[idx] + (1.f - a) * x[idx] + res[idx];
  }
  float mu = wsum(t[0] + t[1] + t[2] + t[3]) * (1.f / 128.f);
  float var = 0.f;
#pragma unroll
  for (int j = 0; j < 4; j++) {
    float d = t[j] - mu;
    var += d * d;
  }
  var = wsum(var) * (1.f / 128.f);
  float inv = rsqrtf(var + 1e-5f);
#pragma unroll
  for (int j = 0; j < 4; j++) {
    int c = j * 32 + lane;
    float v = (t[j] - mu) * inv * g[c] + bb[c];
    y[(long)w * 128 + c] = fmaxf(v, 0.f);
  }
}

__global__ void ln_kernel(const float* __restrict__ src, float* dst,
                          const float* __restrict__ g,
                          const float* __restrict__ bb, int rows) {
  int w = (blockIdx.x * 256 + threadIdx.x) >> 5;
  int lane = threadIdx.x & 31;
  if (w >= rows) return;
  float t[4];
#pragma unroll
  for (int j = 0; j < 4; j++) t[j] = src[(long)w * 128 + j * 32 + lane];
  float mu = wsum(t[0] + t[1] + t[2] + t[3]) * (1.f / 128.f);
  float var = 0.f;
#pragma unroll
  for (int j = 0; j < 4; j++) {
    float d = t[j] - mu;
    var += d * d;
  }
  var = wsum(var) * (1.f / 128.f);
  float inv = rsqrtf(var + 1e-5f);
#pragma unroll
  for (int j = 0; j < 4; j++) {
    int c = j * 32 + lane;
    dst[(long)w * 128 + c] = (t[j] - mu) * inv * g[c] + bb[c];
  }
}

// fused[n,:] = LN(mean_i proj[n*3+i,:])
__global__ void mean_ln_kernel(const float* __restrict__ proj,
                               const float* __restrict__ g,
                               const float* __restrict__ bb, float* dst,
                               int rows) {
  int w = (blockIdx.x * 256 + threadIdx.x) >> 5;
  int lane = threadIdx.x & 31;
  if (w >= rows) return;
  float t[4];
#pragma unroll
  for (int j = 0; j < 4; j++) {
    int c = j * 32 + lane;
    const float* p = proj + (long)w * 3 * 128 + c;
    t[j] = (p[0] + p[128] + p[256]) * (1.f / 3.f);
  }
  float mu = wsum(t[0] + t[1] + t[2] + t[3]) * (1.f / 128.f);
  float var = 0.f;
#pragma unroll
  for (int j = 0; j < 4; j++) {
    float d = t[j] - mu;
    var += d * d;
  }
  var = wsum(var) * (1.f / 128.f);
  float inv = rsqrtf(var + 1e-5f);
#pragma unroll
  for (int j = 0; j < 4; j++) {
    int c = j * 32 + lane;
    dst[(long)w * 128 + c] = (t[j] - mu) * inv * g[c] + bb[c];
  }
}

// out = LN(x + gate*t), writes final output rows
__global__ void gate_mul_ln_kernel(const float* __restrict__ x,
                                   const float* __restrict__ gate,
                                   const float* __restrict__ tt,
                                   const float* __restrict__ g,
                                   const float* __restrict__ bb, float* dst,
                                   int rows) {
  int w = (blockIdx.x * 256 + threadIdx.x) >> 5;
  int lane = threadIdx.x & 31;
  if (w >= rows) return;
  float t[4];
#pragma unroll
  for (int j = 0; j < 4; j++) {
    long idx = (long)w * 128 + j * 32 + lane;
    t[j] = x[idx] + gate[idx] * tt[idx];
  }
  float mu = wsum(t[0] + t[1] + t[2] + t[3]) * (1.f / 128.f);
  float var = 0.f;
#pragma unroll
  for (int j = 0; j < 4; j++) {
    float d = t[j] - mu;
    var += d * d;
  }
  var = wsum(var) * (1.f / 128.f);
  float inv = rsqrtf(var + 1e-5f);
#pragma unroll
  for (int j = 0; j < 4; j++) {
    int c = j * 32 + lane;
    dst[(long)w * 128 + c] = (t[j] - mu) * inv * g[c] + bb[c];
  }
}

// 3-token attention per (node, head): wave = one (n,h); lane = head dim.
__global__ void fusion_attn_kernel(const float* __restrict__ qkv, float* out,
                                   int N) {
  int w = (blockIdx.x * 256 + threadIdx.x) >> 5;
  int lane = threadIdx.x & 31;
  if (w >= N * 4) return;
  int n = w >> 2, h = w & 3;
  const float* base = qkv + (long)n * 3 * 384 + h * 32 + lane;
  float q0 = base[0], q1 = base[384], q2 = base[768];
  float k0 = base[128], k1 = base[512], k2 = base[896];
  float v0 = base[256], v1 = base[640], v2 = base[1024];
  const float sc = 0.17677669529663687f;  // 1/sqrt(32)
  float s00 = wsum(q0 * k0) * sc, s01 = wsum(q0 * k1) * sc, s02 = wsum(q0 * k2) * sc;
  float s10 = wsum(q1 * k0) * sc, s11 = wsum(q1 * k1) * sc, s12 = wsum(q1 * k2) * sc;
  float s20 = wsum(q2 * k0) * sc, s21 = wsum(q2 * k1) * sc, s22 = wsum(q2 * k2) * sc;
  float* ob = out + (long)n * 3 * 128 + h * 32 + lane;
  {
    float m = fmaxf(s00, fmaxf(s01, s02));
    float e0 = expf(s00 - m), e1 = expf(s01 - m), e2 = expf(s02 - m);
    ob[0] = (e0 * v0 + e1 * v1 + e2 * v2) / (e0 + e1 + e2);
  }
  {
    float m = fmaxf(s10, fmaxf(s11, s12));
    float e0 = expf(s10 - m), e1 = expf(s11 - m), e2 = expf(s12 - m);
    ob[128] = (e0 * v0 + e1 * v1 + e2 * v2) / (e0 + e1 + e2);
  }
  {
    float m = fmaxf(s20, fmaxf(s21, s22));
    float e0 = expf(s20 - m), e1 = expf(s21 - m), e2 = expf(s22 - m);
    ob[256] = (e0 * v0 + e1 * v1 + e2 * v2) / (e0 + e1 + e2);
  }
}

__global__ void proto_norm_kernel(const float* __restrict__ protos, float* pn) {
  int w = (blockIdx.x * 256 + threadIdx.x) >> 5;
  int lane = threadIdx.x & 31;
  if (w >= NPROTO) return;
  float x[4], ss = 0.f;
#pragma unroll
  for (int j = 0; j < 4; j++) {
    x[j] = protos[(long)w * 128 + j * 32 + lane];
    ss += x[j] * x[j];
  }
  float nrm = sqrtf(wsum(ss)) + 1e-12f;
#pragma unroll
  for (int j = 0; j < 4; j++) pn[(long)w * 128 + j * 32 + lane] = x[j] / nrm;
}

// x_abs = softmax((x/||x||) @ pn^T / TEMP) @ protos   (wave per node)
__global__ void proto_route_kernel(const float* __restrict__ x,
                                   const float* __restrict__ protos,
                                   const float* __restrict__ pn, float* xabs,
                                   int N) {
  int w = (blockIdx.x * 256 + threadIdx.x) >> 5;
  int lane = threadIdx.x & 31;
  if (w >= N) return;
  float xv[4], ss = 0.f;
#pragma unroll
  for (int j = 0; j < 4; j++) {
    xv[j] = x[(long)w * 128 + j * 32 + lane];
    ss += xv[j] * xv[j];
  }
  float inv = 1.f / (sqrtf(wsum(ss)) + 1e-12f);
  float logit[NPROTO];
  for (int p = 0; p < NPROTO; p++) {
    float d = 0.f;
#pragma unroll
    for (int j = 0; j < 4; j++) d += xv[j] * pn[(long)p * 128 + j * 32 + lane];
    logit[p] = wsum(d) * inv * 10.f;  // / TEMP (0.1)
  }
  float m = logit[0];
  for (int p = 1; p < NPROTO; p++) m = fmaxf(m, logit[p]);
  float sim[NPROTO], sum = 0.f;
  for (int p = 0; p < NPROTO; p++) {
    sim[p] = expf(logit[p] - m);
    sum += sim[p];
  }
  float isum = 1.f / sum;
#pragma unroll
  for (int j = 0; j < 4; j++) {
    float acc = 0.f;
    for (int p = 0; p < NPROTO; p++)
      acc += sim[p] * protos[(long)p * 128 + j * 32 + lane];
    xabs[(long)w * 128 + j * 32 + lane] = acc * isum;
  }
}

// ---------------------------------------------------------------------------
extern "C" void kernel_launch(void* const* d_in, const int* in_sizes, int n_in,
                              void* d_out, int out_size, void* d_ws,
                              size_t ws_size, hipStream_t stream) {
  (void)in_sizes; (void)n_in; (void)out_size; (void)ws_size;
  static const int NT_N[4] = {20000, 20000, 20000, 8000};
  static const int NT_OFF[4] = {0, 20000, 40000, 60000};
  static const int E_SRC[6] = {0, 1, 1, 2, 2, 3};
  static const int E_DST[6] = {1, 0, 2, 1, 3, 2};
  const int B0 = 17, F0 = 185, R0 = 203;

  auto F = [&](int i) -> const float* { return (const float*)d_in[i]; };
  auto blkp = [&](int b, int o) -> const float* {
    return (const float*)d_in[B0 + 56 * b + o];
  };

  // ---- workspace bump allocator (fusion phase aliases block scratch) ----
  float* P = (float*)d_ws;
  long off = 0;
  auto alloc = [&](long n) -> float* { float* p = P + off; off += n; return p; };
  float* emb = alloc(NTOT * 128);
  float* st1 = alloc(NTOT * 128);
  float* st2 = alloc(NTOT * 128);
  float* st3 = alloc(NTOT * 128);
  const long scratch0 = off;
  float* kb = alloc(NTOT * 128);
  float* qb = alloc(NTOT * 128);
  float* vb = alloc(NTOT * 128);
  float* msg = alloc(NTOT * 128);
  float* obuf = alloc(NTOT * 128);
  float* krel = alloc((long)E_PER * 128);
  float* vrel = alloc((long)E_PER * 128);
  float* alphab = alloc((long)E_PER * 4);
  unsigned* mxb = (unsigned*)alloc(20000L * 4);
  float* denb = alloc(20000L * 4);
  off = scratch0;  // fusion/refiner phase runs strictly after the blocks
  float* sbuf = alloc(60000L * 128);
  float* qkvb = alloc(60000L * 384);
  float* attno = alloc(60000L * 128);
  float* projb = alloc(60000L * 128);
  float* fusedb = alloc(20000L * 128);
  float* pnb = alloc((long)NPROTO * 128);
  float* xabsb = alloc(20000L * 128);
  float* tb = alloc(20000L * 128);
  float* catb = alloc(20000L * 256);
  float* gateb = alloc(20000L * 128);

  auto fillu = [&](void* p, unsigned v, long n) {
    fill_u32_kernel<<<(int)((n + 255) / 256), 256, 0, stream>>>((unsigned*)p, v, n);
  };
  auto gemm = [&](const float* A, const int* gth, const float* B,
                  const float* bias, float* C, int M, int N, int K, int lda,
                  int ldb, int ldc, int transb, int act) {
    const int tiles_n = N / 16;
    const int nb = (tiles_n % 4 == 0) ? 4 : 2;
    const int blocks = ((M / 16) * (tiles_n / nb) + 7) / 8;
    const size_t sh = (size_t)K * N * sizeof(float);  // <= 320KB WGP LDS
    if (nb == 2)
      gemm_wmma_kernel<0, 0, 2><<<blocks, 256, sh, stream>>>(A, gth, B, bias, C, M, N, K, lda, ldb, ldc);
    else if (transb)
      gemm_wmma_kernel<1, 0, 4><<<blocks, 256, sh, stream>>>(A, gth, B, bias, C, M, N, K, lda, ldb, ldc);
    else if (act == 0)
      gemm_wmma_kernel<0, 0, 4><<<blocks, 256, sh, stream>>>(A, gth, B, bias, C, M, N, K, lda, ldb, ldc);
    else if (act == 1)
      gemm_wmma_kernel<0, 1, 4><<<blocks, 256, sh, stream>>>(A, gth, B, bias, C, M, N, K, lda, ldb, ldc);
    else
      gemm_wmma_kernel<0, 2, 4><<<blocks, 256, sh, stream>>>(A, gth, B, bias, C, M, N, K, lda, ldb, ldc);
  };

  // ---- input projections: relu(x @ W + b); event = embedding table ----
  for (int t = 0; t < 3; t++)
    gemm(F(t), nullptr, F(10 + 2 * t), F(11 + 2 * t),
         emb + (long)NT_OFF[t] * 128, NT_N[t], 128, 300, 300, 128, 128, 0, 1);
  hipMemcpyAsync(emb + (long)NT_OFF[3] * 128, d_in[16], (size_t)8000 * 128 * 4,
                 hipMemcpyDeviceToDevice, stream);

  // ---- 3 HGT blocks ----
  const float* stagein = emb;
  float* stages[3] = {st1, st2, st3};
  for (int b = 0; b < 3; b++) {
    for (int t = 0; t < 4; t++) {
      long o = (long)NT_OFF[t] * 128;
      gemm(stagein + o, nullptr, blkp(b, 20 + 2 * t), blkp(b, 21 + 2 * t), kb + o, NT_N[t], 128, 128, 128, 128, 128, 0, 0);
      gemm(stagein + o, nullptr, blkp(b, 28 + 2 * t), blkp(b, 29 + 2 * t), qb + o, NT_N[t], 128, 128, 128, 128, 128, 0, 0);
      gemm(stagein + o, nullptr, blkp(b, 36 + 2 * t), blkp(b, 37 + 2 * t), vb + o, NT_N[t], 128, 128, 128, 128, 128, 0, 0);
    }
    fillu(msg, 0u, NTOT * 128);
    for (int e = 0; e < 6; e++) {
      const int* sidx = (const int*)d_in[4 + e];
      const int* didx = sidx + E_PER;
      long so = (long)NT_OFF[E_SRC[e]] * 128;
      long dofo = (long)NT_OFF[E_DST[e]] * 128;
      int ndst = NT_N[E_DST[e]];
      const float* a_rel = blkp(b, 2 + 3 * e);
      const float* m_rel = blkp(b, 3 + 3 * e);
      const float* p_rel = blkp(b, 4 + 3 * e);
      // gathered per-head relation GEMMs: [E,32] @ [32,32] via WMMA
      for (int h = 0; h < 4; h++) {
        gemm(kb + so + h * 32, sidx, a_rel + h * 32 * 32, nullptr, krel + h * 32, E_PER, 32, 32, 128, 32, 128, 0, 0);
        gemm(vb + so + h * 32, sidx, m_rel + h * 32 * 32, nullptr, vrel + h * 32, E_PER, 32, 32, 128, 32, 128, 0, 0);
      }
      fillu(mxb, 0x007FFFFFu, (long)ndst * 4);  // flip(-inf)
      fillu(denb, 0u, (long)ndst * 4);
      int thr = E_PER * 4;
      alpha_max_kernel<<<(thr + 255) / 256, 256, 0, stream>>>(qb + dofo, krel, didx, p_rel, mxb, alphab, E_PER);
      softnorm_kernel<<<(thr + 255) / 256, 256, 0, stream>>>(alphab, mxb, denb, didx, E_PER);
      long thr2 = (long)E_PER * 128;
      scatter_kernel<<<(int)((thr2 + 255) / 256), 256, 0, stream>>>(alphab, denb, vrel, didx, msg + dofo, E_PER);
    }
    gelu_kernel<<<(int)((NTOT * 128 + 255) / 256), 256, 0, stream>>>(msg, NTOT * 128);
    for (int t = 0; t < 4; t++) {
      long o = (long)NT_OFF[t] * 128;
      gemm(msg + o, nullptr, blkp(b, 44 + 2 * t), blkp(b, 45 + 2 * t), obuf + o, NT_N[t], 128, 128, 128, 128, 128, 0, 0);
      epilogue_kernel<<<(NT_N[t] + 7) / 8, 256, 0, stream>>>(
          obuf + o, stagein + o, stagein + o, blkp(b, 52 + t), blkp(b, 0), blkp(b, 1), stages[b] + o, NT_N[t]);
    }
    stagein = stages[b];
  }

  // ---- fusion + prototype refiner for pesticide / disease / plant ----
  for (int f = 0; f < 3; f++) {
    const int N = 20000;
    long o = (long)NT_OFF[f] * 128;
    const float* in_w = F(F0 + 6 * f + 0);
    const float* in_b = F(F0 + 6 * f + 1);
    const float* out_w = F(F0 + 6 * f + 2);
    const float* out_b = F(F0 + 6 * f + 3);
    const float* ln_g = F(F0 + 6 * f + 4);
    const float* ln_b = F(F0 + 6 * f + 5);
    for (int i = 0; i < 3; i++)
      copy_strided_kernel<<<(int)(((long)N * 128 + 255) / 256), 256, 0, stream>>>(
          sbuf + i * 128, 384, stages[i] + o, 128, N, 128);
    gemm(sbuf, nullptr, in_w, in_b, qkvb, 3 * N, 384, 128, 128, 128, 384, 1, 0);
    fusion_attn_kernel<<<(N * 4 + 7) / 8, 256, 0, stream>>>(qkvb, attno, N);
    gemm(attno, nullptr, out_w, out_b, projb, 3 * N, 128, 128, 128, 128, 128, 1, 0);
    mean_ln_kernel<<<(N + 7) / 8, 256, 0, stream>>>(projb, ln_g, ln_b, fusedb, N);

    const float* protos = F(R0 + 9 * f + 0);
    const float* pt_w = F(R0 + 9 * f + 1);
    const float* pt_b = F(R0 + 9 * f + 2);
    const float* ptln_g = F(R0 + 9 * f + 3);
    const float* ptln_b = F(R0 + 9 * f + 4);
    const float* gate_w = F(R0 + 9 * f + 5);
    const float* gate_b = F(R0 + 9 * f + 6);
    const float* fn_g = F(R0 + 9 * f + 7);
    const float* fn_b = F(R0 + 9 * f + 8);
    proto_norm_kernel<<<2, 256, 0, stream>>>(protos, pnb);
    proto_route_kernel<<<(N + 7) / 8, 256, 0, stream>>>(fusedb, protos, pnb, xabsb, N);
    gemm(xabsb, nullptr, pt_w, pt_b, tb, N, 128, 128, 128, 128, 128, 0, 1);
    ln_kernel<<<(N + 7) / 8, 256, 0, stream>>>(tb, tb, ptln_g, ptln_b, N);
    copy_strided_kernel<<<(int)(((long)N * 128 + 255) / 256), 256, 0, stream>>>(catb, 256, fusedb, 128, N, 128);
    copy_strided_kernel<<<(int)(((long)N * 128 + 255) / 256), 256, 0, stream>>>(catb + 128, 256, tb, 128, N, 128);
    gemm(catb, nullptr, gate_w, gate_b, gateb, N, 128, 256, 256, 128, 128, 0, 2);
    gate_mul_ln_kernel<<<(N + 7) / 8, 256, 0, stream>>>(
        fusedb, gateb, tb, fn_g, fn_b, (float*)d_out + (long)f * N * 128, N);
  }
}